// HGMN_78477642432877
// MI455X (gfx1250) — compile-verified
//
#include <hip/hip_runtime.h>
#include <hip/hip_bf16.h>

#define N_NODES 50000
#define N_USERC 20000
#define HID 64
#define MEMD 8
#define RREL 5
#define E_TOT 300000
#define ER_REL 60000
#define FEPS 1e-5f
#define SLOPE 0.2f

// WMMA tiling
#define BSTRIDE 160                  // floats per k-pair row (mod 64 words = 32 -> half-waves on disjoint banks)
#define HP 68                        // hs row pad (floats): 272B rows keep float4 stores aligned
#define TILES_EDGE (ER_REL / 16)     // 3750 tiles per relation
#define TILES_SELF (N_NODES / 16)    // 3125 tiles
#define TPW 8                        // tiles per wave
#define NWAVES 8                     // 256-thread block
#define TILES_WG (TPW * NWAVES)      // 64
#define WGS_EDGE ((TILES_EDGE + TILES_WG - 1) / TILES_WG)  // 59
#define WGS_SELF ((TILES_SELF + TILES_WG - 1) / TILES_WG)  // 49

// LDS layout (floats)
#define SH_B   (256 * BSTRIDE)                  // 40960: B' pair-interleaved [k>>1][o*2 + (k&1)]
#define SH_WC  512
#define SH_BC  8
#define SH_WAVE (16 * HP + 128 + 32)            // hs tile + coef + 32 ints of idx = 1248
#define SH_TOTAL (SH_B + SH_WC + SH_BC + NWAVES * SH_WAVE)
#define SMEM_BYTES (SH_TOTAL * 4)               // 205856 bytes (< 320KB WGP LDS)

typedef float v2f __attribute__((ext_vector_type(2)));
typedef float v8f __attribute__((ext_vector_type(8)));

__device__ __forceinline__ void wave_lds_fence() {
  // DS ops are in-order per wave on CDNA5; this guarantees completion + stops
  // compiler reordering around per-wave LDS producer/consumer phases.
  asm volatile("s_wait_dscnt 0x0" ::: "memory");
}
__device__ __forceinline__ float lrelu(float v) { return v > 0.f ? v : SLOPE * v; }
__device__ __forceinline__ void atomic_add_f32(float* p, float v) {
  __hip_atomic_fetch_add(p, v, __ATOMIC_RELAXED, __HIP_MEMORY_SCOPE_AGENT);
}

// ---------------------------------------------------------------------------
// Fused hypernetwork message kernel:
//   msg[e,:] = (coef[e,:] (x) hs[e,:]) @ B'   with B'[m*64+i, o] = Ww[m,o,i]
// edge_mode=1: rows are edges of relation rel; scatter-add msg into out[dst].
// edge_mode=0: identity rows (self loop); direct store msg into out[row].
// ---------------------------------------------------------------------------
__global__ void __launch_bounds__(256)
hgmn_memenc_kernel(const float* __restrict__ x,
                   const int* __restrict__ src, const int* __restrict__ dst,
                   const float* __restrict__ Wc,   // [R][64][8]
                   const float* __restrict__ bc,   // [R][8]
                   const float* __restrict__ Ww,   // [R][8][64][64]
                   float* __restrict__ out,        // [N][64]
                   int tiles_per_rel, int wgs_per_rel, int er, int edge_mode) {
  extern __shared__ float smem[];
  float* Bs  = smem;                 // [256][BSTRIDE]  B'[(k>>1)][o*2 + (k&1)]
  float* WcS = Bs + SH_B;            // [64][8]
  float* bcS = WcS + SH_WC;          // [8]
  const int tid = threadIdx.x;
  const int wave = tid >> 5;
  const int lane = tid & 31;
  float* hsS   = bcS + SH_BC + wave * SH_WAVE;   // [16][HP] per wave
  float* coefS = hsS + 16 * HP;                  // [16][8]
  int*   idxD  = (int*)(coefS + 128);            // [16]
  int*   idxS  = idxD + 16;                      // [16]

  const int rel = blockIdx.x / wgs_per_rel;
  const int wgr = blockIdx.x % wgs_per_rel;
  const float* Wc_r = Wc + rel * 512;
  const float* bc_r = bc + rel * 8;
  const float* Ww_r = Ww + (size_t)rel * 32768;

  // Stage B' into LDS: float4 global reads -> two contiguous b64 LDS stores
  // (pair layout makes {i,i+1} adjacent at Bs[(m*32 + i/2)*BSTRIDE + 2o]).
  for (int w4 = tid * 4; w4 < 32768; w4 += 1024) {
    float4 v = *(const float4*)(Ww_r + w4);
    int m = w4 >> 12, o = (w4 >> 6) & 63, i = w4 & 63;   // i multiple of 4
    float* p = &Bs[(m * 32 + (i >> 1)) * BSTRIDE + o * 2];
    *(v2f*)p = (v2f){v.x, v.y};
    *(v2f*)(p + BSTRIDE) = (v2f){v.z, v.w};
  }
  for (int w = tid; w < 512; w += 256) WcS[w] = Wc_r[w];
  if (tid < 8) bcS[tid] = bc_r[tid];
  __syncthreads();

  const int g = lane >> 4;        // half-wave group (k pair select)
  const int n16 = lane & 15;      // row (A/D-M) / col (B/D-N) within tile
  const int g2 = g * 2;
  const int rr = lane >> 1;       // cooperative-load row
  const int ch = (lane & 1) * 32; // cooperative-load col half

  for (int t = 0; t < TPW; ++t) {
    const int tile = wgr * TILES_WG + wave * TPW + t;
    if (tile >= tiles_per_rel) break;          // per-wave uniform, no barriers below
    const int row0 = rel * er + tile * 16;

    if (edge_mode) {
      if (lane < 16) { idxD[lane] = dst[row0 + lane]; idxS[lane] = src[row0 + lane]; }
      wave_lds_fence();
    }

    // Load h_dst tile into hsS (float4, 16B-aligned rows of HP=68)
    {
      const int grow = edge_mode ? idxD[rr] : (row0 + rr);
      const float4* gp = (const float4*)(x + (size_t)grow * HID + ch);
      float* dp = &hsS[rr * HP + ch];
#pragma unroll
      for (int q = 0; q < 8; ++q) ((float4*)dp)[q] = gp[q];
    }
    wave_lds_fence();

    // coef[e, m] = leaky(h_dst[e] @ Wc + bc); lane computes 4 m's for e=n16
    {
      float a0 = bcS[g * 4 + 0], a1 = bcS[g * 4 + 1];
      float a2 = bcS[g * 4 + 2], a3 = bcS[g * 4 + 3];
      const float* hr = &hsS[n16 * HP];
#pragma unroll 4
      for (int i = 0; i < HID; ++i) {
        float h = hr[i];
        float4 wv = *(const float4*)&WcS[i * 8 + g * 4];   // one b128 LDS read
        a0 += h * wv.x; a1 += h * wv.y; a2 += h * wv.z; a3 += h * wv.w;
      }
      coefS[n16 * 8 + g * 4 + 0] = lrelu(a0);
      coefS[n16 * 8 + g * 4 + 1] = lrelu(a1);
      coefS[n16 * 8 + g * 4 + 2] = lrelu(a2);
      coefS[n16 * 8 + g * 4 + 3] = lrelu(a3);
    }
    wave_lds_fence();
    float coef[8];
#pragma unroll
    for (int m = 0; m < MEMD; ++m) coef[m] = coefS[n16 * 8 + m];

    if (edge_mode) {  // overwrite hsS with h_src tile (identity: hs == hd)
      const int grow = idxS[rr];
      const float4* gp = (const float4*)(x + (size_t)grow * HID + ch);
      float* dp = &hsS[rr * HP + ch];
#pragma unroll
      for (int q = 0; q < 8; ++q) ((float4*)dp)[q] = gp[q];
      wave_lds_fence();
    }

    // Main GEMM: [16 x 512] x [512 x 64] via v_wmma_f32_16x16x4_f32.
    // A fragment: one ds_load_b64 + v_pk_mul_f32; B fragments: one ds_load_b64 each.
    v8f c0 = {0,0,0,0,0,0,0,0}, c1 = c0, c2 = c0, c3 = c0;
    const float* hrow = &hsS[n16 * HP];
#pragma unroll
    for (int m = 0; m < MEMD; ++m) {
      const float cf = coef[m];
      v2f cfv = {cf, cf};
#pragma unroll 4
      for (int k2 = 0; k2 < 16; ++k2) {
        const int i = k2 * 4 + g2;                    // even -> 8B-aligned pair
        v2f hv = *(const v2f*)(hrow + i);
        v2f a = cfv * hv;
        const int p0 = m * 32 + k2 * 2 + g;           // k-pair row (k0>>1)
        const float* bp = &Bs[p0 * BSTRIDE + n16 * 2];
        v2f b0 = *(const v2f*)(bp +  0);
        v2f b1 = *(const v2f*)(bp + 32);
        v2f b2 = *(const v2f*)(bp + 64);
        v2f b3 = *(const v2f*)(bp + 96);
        c0 = __builtin_amdgcn_wmma_f32_16x16x4_f32(false, a, false, b0, (short)0, c0, false, false);
        c1 = __builtin_amdgcn_wmma_f32_16x16x4_f32(false, a, false, b1, (short)0, c1, false, false);
        c2 = __builtin_amdgcn_wmma_f32_16x16x4_f32(false, a, false, b2, (short)0, c2, false, false);
        c3 = __builtin_amdgcn_wmma_f32_16x16x4_f32(false, a, false, b3, (short)0, c3, false, false);
      }
    }

    // Epilogue: D layout -> vgpr j holds row (j + 8*g), col (16*t + n16)
    if (edge_mode) {
#pragma unroll
      for (int j = 0; j < 8; ++j) {
        const int grow = idxD[j + 8 * g];
        float* op = out + (size_t)grow * HID + n16;
        atomic_add_f32(op +  0, c0[j]);
        atomic_add_f32(op + 16, c1[j]);
        atomic_add_f32(op + 32, c2[j]);
        atomic_add_f32(op + 48, c3[j]);
      }
    } else {
#pragma unroll
      for (int j = 0; j < 8; ++j) {
        float* op = out + (size_t)(row0 + j + 8 * g) * HID + n16;
        op[0] = c0[j]; op[16] = c1[j]; op[32] = c2[j]; op[48] = c3[j];
      }
    }
  }
}

// ---------------------------------------------------------------------------
__global__ void hgmn_zero_kernel(float* __restrict__ p, long n) {
  long i = (long)blockIdx.x * blockDim.x + threadIdx.x;
  long stride = (long)gridDim.x * blockDim.x;
  for (; i < n; i += stride) p[i] = 0.f;
}

__global__ void hgmn_indeg_kernel(const int* __restrict__ dst, float* __restrict__ cnt, int e) {
  int i = blockIdx.x * blockDim.x + threadIdx.x;
  if (i < e) atomic_add_f32(&cnt[dst[i]], 1.f);
}

__global__ void hgmn_node_update_kernel(const float* __restrict__ acc, const float* __restrict__ cnt,
                                        const float* __restrict__ selfmsg,
                                        const float* __restrict__ lnw, const float* __restrict__ lnb,
                                        const float* __restrict__ hb,
                                        float* __restrict__ xout, int n) {
  int v = blockIdx.x * blockDim.x + threadIdx.x;
  if (v >= n) return;
  const float* ar = acc + (size_t)v * HID;
  float c = cnt[v]; c = c < 1.f ? 1.f : c;
  float ic = 1.f / c;
  float s = 0.f, s2 = 0.f;
  for (int i = 0; i < HID; ++i) { float t = ar[i] * ic; s += t; s2 += t * t; }
  float mean = s * (1.f / HID);
  float var = s2 * (1.f / HID) - mean * mean;
  float rstd = rsqrtf(var + FEPS);
  const float* sm = selfmsg + (size_t)v * HID;
  float* xo = xout + (size_t)v * HID;
  for (int i = 0; i < HID; ++i) {
    float t = (ar[i] * ic - mean) * rstd * lnw[i] + lnb[i] + hb[i] + sm[i];
    xo[i] = lrelu(t);
  }
}

__global__ void hgmn_concat_ln_kernel(const float* __restrict__ x0, const float* __restrict__ x1,
                                      const float* __restrict__ x2,
                                      const float* __restrict__ w, const float* __restrict__ b,
                                      float* __restrict__ out, int n) {
  int v = blockIdx.x * blockDim.x + threadIdx.x;
  if (v >= n) return;
  const float* xs[3] = {x0 + (size_t)v * HID, x1 + (size_t)v * HID, x2 + (size_t)v * HID};
  float s = 0.f, s2 = 0.f;
  for (int p = 0; p < 3; ++p)
    for (int i = 0; i < HID; ++i) { float t = xs[p][i]; s += t; s2 += t * t; }
  float mean = s * (1.f / 192.f);
  float var = s2 * (1.f / 192.f) - mean * mean;
  float rstd = rsqrtf(var + FEPS);
  float* o = out + (size_t)v * 192;
  for (int p = 0; p < 3; ++p)
    for (int i = 0; i < HID; ++i) {
      int j = p * HID + i;
      o[j] = (xs[p][i] - mean) * rstd * w[j] + b[j];
    }
}

__global__ void hgmn_pool_init_kernel(const float* __restrict__ xcat, float* __restrict__ pool,
                                      float* __restrict__ pcnt, int nuser) {
  long t = (long)blockIdx.x * blockDim.x + threadIdx.x;
  long n = (long)nuser * 192;
  if (t >= n) return;
  pool[t] = xcat[t];                 // identity self-loop term
  if ((t % 192) == 0) pcnt[t / 192] = 1.f;
}

__global__ void hgmn_pool_edge_kernel(const int* __restrict__ src, const int* __restrict__ dst,
                                      const float* __restrict__ xcat, float* __restrict__ pool,
                                      float* __restrict__ pcnt, int er) {
  long t = (long)blockIdx.x * blockDim.x + threadIdx.x;
  if (t >= (long)er * 12) return;
  int e = (int)(t / 12);
  int j0 = (int)(t % 12) * 16;
  int d = dst[e], s = src[e];
  if (j0 == 0) atomic_add_f32(&pcnt[d], 1.f);
  const float* xr = xcat + (size_t)s * 192 + j0;
  float* pr = pool + (size_t)d * 192 + j0;
#pragma unroll
  for (int j = 0; j < 16; ++j) atomic_add_f32(&pr[j], xr[j]);
}

__global__ void hgmn_pool_norm_kernel(float* __restrict__ pool, const float* __restrict__ pcnt,
                                      int nuser) {
  long t = (long)blockIdx.x * blockDim.x + threadIdx.x;
  long n = (long)nuser * 192;
  if (t >= n) return;
  pool[t] /= pcnt[t / 192];
}

// ---------------------------------------------------------------------------
extern "C" void kernel_launch(void* const* d_in, const int* in_sizes, int n_in,
                              void* d_out, int out_size, void* d_ws, size_t ws_size,
                              hipStream_t stream) {
  const float* emb     = (const float*)d_in[0];
  const float* Wc_rel  = (const float*)d_in[1];
  const float* bc_rel  = (const float*)d_in[2];
  const float* Ww_rel  = (const float*)d_in[3];
  const float* Wc_node = (const float*)d_in[4];
  const float* bc_node = (const float*)d_in[5];
  const float* Ww_node = (const float*)d_in[6];
  const float* h_bias  = (const float*)d_in[7];
  const float* ln_w    = (const float*)d_in[8];
  const float* ln_b    = (const float*)d_in[9];
  const float* norm_w  = (const float*)d_in[10];
  const float* norm_b  = (const float*)d_in[11];
  const int*   src     = (const int*)d_in[12];
  const int*   dst     = (const int*)d_in[13];

  float* ws   = (float*)d_ws;
  float* x1   = ws;
  float* x2   = x1 + (size_t)N_NODES * HID;
  float* acc  = x2 + (size_t)N_NODES * HID;
  float* self = acc + (size_t)N_NODES * HID;
  float* cnt  = self + (size_t)N_NODES * HID;
  float* pcnt = cnt + N_NODES;

  // in-degree (computed once, shared across layers)
  hgmn_zero_kernel<<<256, 256, 0, stream>>>(cnt, (long)N_NODES);
  hgmn_indeg_kernel<<<(E_TOT + 255) / 256, 256, 0, stream>>>(dst, cnt, E_TOT);

  const float* xcur = emb;
  float* xbuf[2] = {x1, x2};
  for (int l = 0; l < 2; ++l) {
    hgmn_zero_kernel<<<512, 256, 0, stream>>>(acc, (long)N_NODES * HID);
    // per-relation edge messages, scatter-add into acc (fused segment_sum)
    hgmn_memenc_kernel<<<RREL * WGS_EDGE, 256, SMEM_BYTES, stream>>>(
        xcur, src, dst,
        Wc_rel + (size_t)l * RREL * 512, bc_rel + (size_t)l * RREL * 8,
        Ww_rel + (size_t)l * RREL * 32768, acc, TILES_EDGE, WGS_EDGE, ER_REL, 1);
    // self-loop hypernetwork term (identity rows, direct store)
    hgmn_memenc_kernel<<<WGS_SELF, 256, SMEM_BYTES, stream>>>(
        xcur, nullptr, nullptr,
        Wc_node + (size_t)l * 512, bc_node + (size_t)l * 8,
        Ww_node + (size_t)l * 32768, self, TILES_SELF, WGS_SELF, 0, 0);
    // x = leaky( LN(acc/cnt)*w + b + h_bias + self )
    hgmn_node_update_kernel<<<(N_NODES + 127) / 128, 128, 0, stream>>>(
        acc, cnt, self, ln_w + l * HID, ln_b + l * HID, h_bias + l * HID, xbuf[l], N_NODES);
    xcur = xbuf[l];
  }

  float* xcat = (float*)d_out;                 // [N, 192]
  float* pool = xcat + (size_t)N_NODES * 192;  // [N_USER, 192]
  hgmn_concat_ln_kernel<<<(N_NODES + 127) / 128, 128, 0, stream>>>(
      emb, x1, x2, norm_w, norm_b, xcat, N_NODES);
  hgmn_pool_init_kernel<<<((long)N_USERC * 192 + 255) / 256, 256, 0, stream>>>(
      xcat, pool, pcnt, N_USERC);
  hgmn_pool_edge_kernel<<<((long)ER_REL * 12 + 255) / 256, 256, 0, stream>>>(
      src, dst, xcat, pool, pcnt, ER_REL);
  hgmn_pool_norm_kernel<<<((long)N_USERC * 192 + 255) / 256, 256, 0, stream>>>(
      pool, pcnt, N_USERC);
}